// HD_loss_62551903699068
// MI455X (gfx1250) — compile-verified
//
#include <hip/hip_runtime.h>
#include <hip/hip_bf16.h>

// ---------------------------------------------------------------------------
// HD loss (Hausdorff DT loss), POWER=2, for x:(2,4,64,64,64) f32, y:(2,64,64,64) int
//   y_oh = onehot(y); x_oh = onehot(argmax_c x)
//   dtm  = exact squared EDT (3 separable brute-force min-plus passes, L=64),
//          zeroed per (b,c) when the class has no foreground
//   out  = mean over (b, c=1..3, voxels) of (x_oh-y_oh)^2 * (gt_dtm^2 + seg_dtm^2)
// ---------------------------------------------------------------------------

#define NB    2
#define NC    4
#define DD    64
#define HH    64
#define WW    64
#define NVOX  (DD*HH*WW)          // 262144 per (b,c) volume
#define NVOL  (NB*NC)             // 8 volumes per mask
#define NPOS  (NB*NVOX)           // 524288 spatial positions
#define BIGV  12288.0f            // D*D+H*H+W*W upper bound (matches reference)
#define RED_BLOCKS 1024           // map_reduce grid; 1024*256*2 == NPOS exactly
#define MEAN_DEN   1572864.0f     // B*(C-1)*D*H*W

typedef __attribute__((ext_vector_type(2))) float v2f;
typedef __attribute__((ext_vector_type(8))) float v8f;

// ---------------------------------------------------------------- flag reset
__global__ void zero_flags_kernel(int* flags) {
    flags[0] = 0;   // y-mask per-(b,c) presence bitmask (bit b*4+c)
    flags[1] = 0;   // x-mask presence bitmask
}

// ------------------------------------------------------- init g = onehot?BIG:0
// G layout: [0 .. 8*NVOX)     = y-mask volumes (b-major, c-minor)
//           [8*NVOX..16*NVOX) = x-mask (argmax) volumes
__global__ void init_kernel(const float* __restrict__ x,
                            const int*   __restrict__ y,
                            float* __restrict__ G,
                            int*   __restrict__ flags) {
    __shared__ int sfy, sfx;
    if (threadIdx.x == 0) { sfy = 0; sfx = 0; }
    __syncthreads();

    int pos = blockIdx.x * blockDim.x + threadIdx.x;       // < NPOS (exact grid)
    int b   = pos / NVOX;
    int sp  = pos - b * NVOX;

    int yc = y[pos];                                       // 0..3

    const float* xb = x + (b * NC) * NVOX + sp;            // argmax over channels
    float best = xb[0];
    int   xa   = 0;
    #pragma unroll
    for (int c = 1; c < NC; ++c) {
        float v = xb[c * NVOX];
        if (v > best) { best = v; xa = c; }                // first-max tie rule
    }

    float* Gy = G;
    float* Gx = G + 8 * NVOX;
    #pragma unroll
    for (int c = 0; c < NC; ++c) {
        int idx = (b * NC + c) * NVOX + sp;
        Gy[idx] = (yc == c) ? BIGV : 0.0f;
        Gx[idx] = (xa == c) ? BIGV : 0.0f;
    }

    atomicOr(&sfy, 1 << (b * NC + yc));
    atomicOr(&sfx, 1 << (b * NC + xa));
    __syncthreads();
    if (threadIdx.x == 0) {
        atomicOr(&flags[0], sfy);
        atomicOr(&flags[1], sfx);
    }
}

// --------------------------------------------- one separable min-plus EDT pass
// axis: 0 = W (stride 1), 1 = H (stride 64), 2 = D (stride 4096).
// One 64-thread group per line; 4 lines per 256-thread block; in-place safe
// (each line is fully owned by its group; groups barrier between load & store).
// The global->LDS staging copy goes through the gfx1250 async-to-LDS path
// (data never needs to touch VGPRs before LDS); falls back to a plain LDS
// store if the builtin is absent.
__global__ void edt_pass_kernel(float* __restrict__ G, int axis) {
    __shared__ float sline[4][64];
    const int grp = threadIdx.x >> 6;        // 0..3
    const int i   = threadIdx.x & 63;        // position along the line
    const int L   = blockIdx.x * 4 + grp;    // global line id, < 16*4096
    const int v   = L >> 12;                 // volume 0..15
    const int r   = L & 4095;                // line within volume

    int base, stride;
    if (axis == 0)      { base = v * NVOX + (r << 6);                    stride = 1;    }
    else if (axis == 1) { base = v * NVOX + ((r >> 6) << 12) + (r & 63); stride = 64;   }
    else                { base = v * NVOX + r;                           stride = 4096; }

    float* gp = G + base + i * stride;

#if defined(__HIP_DEVICE_COMPILE__) && \
    __has_builtin(__builtin_amdgcn_global_load_async_to_lds_b32)
    {
        typedef __attribute__((address_space(1))) int GASI;   // global int*
        typedef __attribute__((address_space(3))) int LASI;   // shared int*
        __builtin_amdgcn_global_load_async_to_lds_b32(
            (GASI*)gp, (LASI*)&sline[grp][i], /*offset=*/0, /*cpol=*/0);
#if __has_builtin(__builtin_amdgcn_s_wait_asynccnt)
        __builtin_amdgcn_s_wait_asynccnt(0);
#else
        asm volatile("s_wait_asynccnt 0x0" ::: "memory");
#endif
    }
#else
    sline[grp][i] = *gp;
#endif
    __syncthreads();

    float m = 3.4e38f;
    #pragma unroll 8
    for (int j = 0; j < 64; ++j) {           // LDS broadcast read per j
        float d = (float)(i - j);
        m = fminf(m, sline[grp][j] + d * d);
    }
    __syncthreads();
    *gp = m;
}

// ------------------------------------------- per-voxel term of the loss sum
__device__ __forceinline__ float voxel_term(const float* __restrict__ x,
                                            const int*   __restrict__ y,
                                            const float* __restrict__ Gy,
                                            const float* __restrict__ Gx,
                                            int flagY, int flagX, int pos) {
    int b  = pos / NVOX;
    int sp = pos - b * NVOX;
    int yc = y[pos];

    const float* xb = x + (b * NC) * NVOX + sp;
    float best = xb[0];
    int   xa   = 0;
    #pragma unroll
    for (int c = 1; c < NC; ++c) {
        float v = xb[c * NVOX];
        if (v > best) { best = v; xa = c; }
    }

    float acc = 0.0f;
    #pragma unroll
    for (int c = 1; c < NC; ++c) {           // class 0 excluded by [:,1:]
        float delta2 = (float)((xa == c) != (yc == c));   // (x_oh - y_oh)^2
        int   idx    = (b * NC + c) * NVOX + sp;
        int   bit    = b * NC + c;
        float gy = ((flagY >> bit) & 1) ? Gy[idx] : 0.0f; // any_fg zeroing
        float gx = ((flagX >> bit) & 1) ? Gx[idx] : 0.0f;
        float ty = sqrtf(gy);                             // mimic sqrt(g)**2
        float tx = sqrtf(gx);
        acc += delta2 * (ty * ty + tx * tx);
    }
    return acc;
}

// ---------------------- map + reduce; wave reduction via V_WMMA_F32_16X16X4_F32
// Exactly 2 positions per thread (uniform trip count -> EXEC all-1s at WMMA).
// A(16x4) holds the 64 per-wave partials (2 per lane); B = ones; D rows carry
// A row-sums; per-lane sum of the 8 D VGPRs + shfl_xor(16) = full wave sum.
__global__ void map_reduce_kernel(const float* __restrict__ x,
                                  const int*   __restrict__ y,
                                  const float* __restrict__ G,
                                  const int*   __restrict__ flags,
                                  float* __restrict__ blocksums) {
    const float* Gy = G;
    const float* Gx = G + 8 * NVOX;
    const int flagY = flags[0];
    const int flagX = flags[1];

    const int gid = blockIdx.x * blockDim.x + threadIdx.x;   // grid = 262144
    float p0 = voxel_term(x, y, Gy, Gx, flagY, flagX, gid);
    float p1 = voxel_term(x, y, Gy, Gx, flagY, flagX, gid + RED_BLOCKS * 256);

    // Wave32 reduction through the matrix unit (f32 accumulate).
    v2f a;  a.x = p0;   a.y = p1;
    v2f bm; bm.x = 1.0f; bm.y = 1.0f;
    v8f c = {};
    c = __builtin_amdgcn_wmma_f32_16x16x4_f32(
            /*neg_a=*/false, a, /*neg_b=*/false, bm,
            /*c_mod=*/(short)0, c, /*reuse_a=*/false, /*reuse_b=*/false);
    float s8 = c[0] + c[1] + c[2] + c[3] + c[4] + c[5] + c[6] + c[7];
    float wave_sum = s8 + __shfl_xor(s8, 16, 32);            // halves M=0..7 / 8..15

    __shared__ float swave[8];                               // 8 waves / 256 thr
    const int lane = threadIdx.x & 31;
    const int wid  = threadIdx.x >> 5;
    if (lane == 0) swave[wid] = wave_sum;
    __syncthreads();
    if (threadIdx.x == 0) {
        float t = 0.0f;
        #pragma unroll
        for (int w = 0; w < 8; ++w) t += swave[w];           // fixed order
        blocksums[blockIdx.x] = t;
    }
}

// ----------------------------------------------- deterministic final reduce
__global__ void final_reduce_kernel(const float* __restrict__ blocksums,
                                    float* __restrict__ out) {
    __shared__ float s[256];
    float v = 0.0f;
    for (int k = threadIdx.x; k < RED_BLOCKS; k += 256) v += blocksums[k];
    s[threadIdx.x] = v;
    __syncthreads();
    for (int off = 128; off > 0; off >>= 1) {
        if (threadIdx.x < off) s[threadIdx.x] += s[threadIdx.x + off];
        __syncthreads();
    }
    if (threadIdx.x == 0) out[0] = s[0] * (1.0f / MEAN_DEN);
}

// ---------------------------------------------------------------------------
extern "C" void kernel_launch(void* const* d_in, const int* in_sizes, int n_in,
                              void* d_out, int out_size, void* d_ws, size_t ws_size,
                              hipStream_t stream) {
    const float* x = (const float*)d_in[0];   // (2,4,64,64,64) f32
    const int*   y = (const int*)d_in[1];     // (2,64,64,64) int

    // Workspace: 16 MB of EDT volumes + flags + per-block sums.
    float* G         = (float*)d_ws;                       // 16*NVOX floats
    int*   flags     = (int*)(G + 16 * NVOX);              // 2 ints
    float* blocksums = (float*)(flags + 8);                // RED_BLOCKS floats

    zero_flags_kernel<<<1, 1, 0, stream>>>(flags);

    init_kernel<<<NPOS / 256, 256, 0, stream>>>(x, y, G, flags);

    // Three separable EDT passes over all 16 volumes (in place).
    const int nLineBlocks = (16 * 4096) / 4;               // 16384
    edt_pass_kernel<<<nLineBlocks, 256, 0, stream>>>(G, 0);
    edt_pass_kernel<<<nLineBlocks, 256, 0, stream>>>(G, 1);
    edt_pass_kernel<<<nLineBlocks, 256, 0, stream>>>(G, 2);

    map_reduce_kernel<<<RED_BLOCKS, 256, 0, stream>>>(x, y, G, flags, blocksums);
    final_reduce_kernel<<<1, 256, 0, stream>>>(blocksums, (float*)d_out);
}